// FeatureNormalizedMSE_67869073212117
// MI455X (gfx1250) — compile-verified
//
#include <hip/hip_runtime.h>

// ---------------------------------------------------------------------------
// FeatureNormalizedMSE for MI455X (gfx1250, wave32).
//
// Memory-bound: 256 MiB of f32 streamed once -> ~11.5 us at 23.3 TB/s.
// Hot loop: coalesced 128-bit non-temporal loads (streaming data, keep L2 for
// the feature-weight vector, which is 64 KiB and re-read by every block).
// Cross-lane reductions use V_WMMA_F32_16X16X4_F32 with a ones A-matrix:
//   D[m][n] = sum_k B[k][n]  ==> column-n sum = v[n] + v[n+16]  (32 -> 16 fold)
// followed by 4 shfl_xor steps. Exact fp32, deterministic order.
// ---------------------------------------------------------------------------

typedef float v2f __attribute__((ext_vector_type(2)));
typedef float v4f __attribute__((ext_vector_type(4)));
typedef float v8f __attribute__((ext_vector_type(8)));

__device__ __forceinline__ float wave_reduce_sum_wmma(float v) {
    // A = ones(16x4): every lane supplies two A slots -> set both to 1.0.
    v2f a; a.x = 1.0f; a.y = 1.0f;
    // B(4x16): lane L contributes its partial into column (L & 15).
    v2f b; b.x = v;    b.y = 0.0f;
    v8f c = {0.f, 0.f, 0.f, 0.f, 0.f, 0.f, 0.f, 0.f};
    // D = A*B + 0 : every row of D equals the per-column sum of B.
    v8f d = __builtin_amdgcn_wmma_f32_16x16x4_f32(
        /*neg_a=*/false, a, /*neg_b=*/false, b,
        /*c_mod=*/(short)0, c, /*reuse_a=*/false, /*reuse_b=*/false);
    float s = d[0];                 // s(lane) = v[lane&15] + v[(lane&15)+16]
    s += __shfl_xor(s, 1, 32);      // fold the 16 remaining distinct values
    s += __shfl_xor(s, 2, 32);
    s += __shfl_xor(s, 4, 32);
    s += __shfl_xor(s, 8, 32);
    return s;                       // all 32 lanes hold the full wave sum
}

// Block of 256 threads = 8 wave32s.
__device__ __forceinline__ float block_reduce_sum_256(float v, float* lds8) {
    float s = wave_reduce_sum_wmma(v);
    const int wid  = threadIdx.x >> 5;
    const int lane = threadIdx.x & 31;
    if (lane == 0) lds8[wid] = s;
    __syncthreads();
    float total = 0.f;
#pragma unroll
    for (int i = 0; i < 8; ++i) total += lds8[i];   // fixed order, all threads
    __syncthreads();                                // lds8 reusable afterwards
    return total;
}

// One workgroup per sample row b: streams 2 x 64 KiB (output/target row) and
// the cached 64 KiB feature-weight vector; emits per_sample[b] = sum(r^2)/cnt.
__global__ __launch_bounds__(256)
void fnmse_per_sample_kernel(const float* __restrict__ outp,
                             const float* __restrict__ tgtp,
                             const float* __restrict__ fwp,
                             float* __restrict__ per_sample,
                             int F) {
    __shared__ float lds8[8];
    const int b = blockIdx.x;
    const v4f* __restrict__ o4 = reinterpret_cast<const v4f*>(outp + (size_t)b * F);
    const v4f* __restrict__ t4 = reinterpret_cast<const v4f*>(tgtp + (size_t)b * F);
    const v4f* __restrict__ w4 = reinterpret_cast<const v4f*>(fwp);

    float sq = 0.f, ct = 0.f;
    const int n4 = F >> 2;                           // 4096 float4 per row
#pragma unroll 4
    for (int i = threadIdx.x; i < n4; i += 256) {
        // Streaming data: non-temporal (keep L2 capacity for fw reuse).
        v4f o = __builtin_nontemporal_load(&o4[i]);
        v4f t = __builtin_nontemporal_load(&t4[i]);
        v4f w = w4[i];                               // cache-friendly reuse
#pragma unroll
        for (int k = 0; k < 4; ++k) {
            float tk = t[k];
            bool  m  = (tk == tk);                   // !isnan
            float r  = m ? (tk - o[k]) * w[k] : 0.f; // avoids NaN propagation
            sq = fmaf(r, r, sq);
            ct += m ? 1.f : 0.f;
        }
    }
    float SQ = block_reduce_sum_256(sq, lds8);
    float CT = block_reduce_sum_256(ct, lds8);
    if (threadIdx.x == 0) per_sample[b] = SQ / CT;
}

// Single block: fold the B per-sample values into one scalar, fixed order.
__global__ __launch_bounds__(256)
void fnmse_final_kernel(const float* __restrict__ per_sample,
                        float* __restrict__ out,
                        int B) {
    __shared__ float lds8[8];
    float s = 0.f;
    for (int i = threadIdx.x; i < B; i += 256) s += per_sample[i];
    float T = block_reduce_sum_256(s, lds8);
    if (threadIdx.x == 0) out[0] = T;
}

extern "C" void kernel_launch(void* const* d_in, const int* in_sizes, int n_in,
                              void* d_out, int out_size, void* d_ws, size_t ws_size,
                              hipStream_t stream) {
    // setup_inputs order: output, target, e_exp, sample_weight, feature_weight
    const float* outp = (const float*)d_in[0];
    const float* tgtp = (const float*)d_in[1];
    const float* fwp  = (const float*)d_in[4];   // [F]
    float* ws         = (float*)d_ws;            // per_sample[B] scratch
    float* res        = (float*)d_out;           // single f32 result

    const int F = in_sizes[4];                   // 16384
    const int B = in_sizes[0] / F;               // 2048

    fnmse_per_sample_kernel<<<B, 256, 0, stream>>>(outp, tgtp, fwp, ws, F);
    fnmse_final_kernel<<<1, 256, 0, stream>>>(ws, res, B);
}